// STDPLinear_77687368450061
// MI455X (gfx1250) — compile-verified
//
#include <hip/hip_runtime.h>

#define IN_F 8192
#define OUT_F 8192
#define BETA 0.99f
#define THRESHOLD 1.0f
#define RESET 0.8f
#define TAU 20.0f
#define A_POS 0.005f
#define A_NEG 0.005f

typedef __attribute__((ext_vector_type(2))) float v2f;
typedef __attribute__((ext_vector_type(4))) float v4f;
typedef __attribute__((ext_vector_type(8))) float v8f;

// ---------------------------------------------------------------------------
// Kernel 1: weighted = W @ spikes via fp32 WMMA (v_wmma_f32_16x16x4_f32).
// Workgroup = 256 threads = 8 waves. Each WG owns a 16-row block of W.
// Wave w handles K-slice [w*1024, (w+1)*1024). B is the spike slice broadcast
// across all 16 columns, so every column of D equals the row dot-product.
// A layout (ISA 7.12.2, 32-bit A 16x4): lanes 0-15 -> M=lane, VGPR0/1 = K0/K1;
// lanes 16-31 -> M=lane-16, VGPR0/1 = K2/K3.
// Wave partials are reduced deterministically through LDS.
// ---------------------------------------------------------------------------
__global__ __launch_bounds__(256) void gevm_wmma_kernel(
    const float* __restrict__ W, const float* __restrict__ spikes,
    float* __restrict__ weighted) {
  __shared__ float s_sp[IN_F];
  __shared__ float s_part[8 * 16];

  const int tid = threadIdx.x;
  // Stage the full spike vector (32 KB) into LDS, coalesced b128 loads.
  for (int t = tid; t < IN_F / 4; t += 256) {
    ((v4f*)s_sp)[t] = ((const v4f*)spikes)[t];
  }
  __syncthreads();

  const int wave = tid >> 5;
  const int lane = tid & 31;
  const int rowBase = blockIdx.x * 16;
  const int koff = (lane >> 4) * 2;          // K sub-offset for this half-wave
  const int kbase = wave * (IN_F / 8);       // 1024-wide K slice per wave
  const float* wrow = W + (size_t)(rowBase + (lane & 15)) * IN_F + koff;

  v8f c = {};
#pragma unroll 4
  for (int k = kbase; k < kbase + IN_F / 8; k += 4) {
    v2f a = *(const v2f*)(wrow + k);         // A: K = k+koff, k+koff+1
    v2f b;
    b.x = s_sp[k + koff];                    // B row K broadcast over N
    b.y = s_sp[k + koff + 1];
    c = __builtin_amdgcn_wmma_f32_16x16x4_f32(
        /*neg_a=*/false, a, /*neg_b=*/false, b,
        /*c_mod=*/(short)0, c, /*reuse_a=*/false, /*reuse_b=*/false);
  }

  // D layout (32-bit C/D 16x16): VGPR r, lanes 0-15 -> M=r (N=lane);
  // lanes 16-31 -> M=8+r. All columns identical, so read column N=0/N=16.
  if (lane == 0) {
#pragma unroll
    for (int r = 0; r < 8; ++r) s_part[wave * 16 + r] = c[r];
  }
  if (lane == 16) {
#pragma unroll
    for (int r = 0; r < 8; ++r) s_part[wave * 16 + 8 + r] = c[r];
  }
  __syncthreads();

  if (tid < 16) {
    float s = 0.f;
#pragma unroll
    for (int w = 0; w < 8; ++w) s += s_part[w * 16 + tid];
    weighted[rowBase + tid] = s;
  }
}

// ---------------------------------------------------------------------------
// Kernel 2: LIF update + delta counters + separable exp tables.
// exp(-+(f-p)/tau) = exp(-+f/tau) * exp(+-p/tau): column factors precomputed
// here (L2-resident, 64 KB), row factors computed per-row in kernel 3.
// ---------------------------------------------------------------------------
__global__ __launch_bounds__(256) void lif_kernel(
    const float* __restrict__ weighted, const float* __restrict__ in_spikes,
    const float* __restrict__ membrane, const float* __restrict__ delta_pre,
    const float* __restrict__ delta_fire, float* __restrict__ out,
    float* __restrict__ colA, float* __restrict__ colB) {
  const int i = blockIdx.x * blockDim.x + threadIdx.x;
  if (i >= OUT_F) return;

  const float mem = membrane[i] * BETA + weighted[i];
  const float spike = (mem > THRESHOLD) ? 1.f : 0.f;
  const float nmem = (spike > 0.f) ? (mem - RESET) : mem;
  const float ndp = (in_spikes[i] > 0.f) ? 0.f : (delta_pre[i] + 1.f);
  const float ndf = (spike > 0.f) ? 0.f : (delta_fire[i] + 1.f);

  out[0 * OUT_F + i] = spike;
  out[1 * OUT_F + i] = nmem;
  out[2 * OUT_F + i] = ndp;
  out[3 * OUT_F + i] = ndf;

  colA[i] = __expf(ndp / TAU);    // exp(+p/tau)  (potentiation column term)
  colB[i] = __expf(-ndp / TAU);   // exp(-p/tau)  (depression column term)
}

// ---------------------------------------------------------------------------
// Kernel 3: new_W[o][i] = W[o][i] + STDP(f_o, p_i). One row per block.
// Streaming float4: NT-load W, NT-store new_W (don't thrash the 192 MB L2
// with the 256 MB write stream). Small tables stay L2-resident.
// ---------------------------------------------------------------------------
__global__ __launch_bounds__(256) void stdp_update_kernel(
    const float* __restrict__ W, const float* __restrict__ out,
    const float* __restrict__ colA, const float* __restrict__ colB,
    float* __restrict__ newW) {
  const int o = blockIdx.x;
  const float f = out[3 * OUT_F + o];              // new_delta_fire[o]
  const float rowP = A_POS * __expf(-f / TAU);     // potentiation row term
  const float rowD = A_NEG * __expf(f / TAU);      // depression row term
  const float* ndp = out + 2 * OUT_F;              // new_delta_pre

  const v4f* W4 = (const v4f*)(W + (size_t)o * IN_F);
  v4f* O4 = (v4f*)(newW + (size_t)o * IN_F);
  const v4f* P4 = (const v4f*)ndp;
  const v4f* CA4 = (const v4f*)colA;
  const v4f* CB4 = (const v4f*)colB;

  for (int t = threadIdx.x; t < IN_F / 4; t += 256) {
    v4f w = __builtin_nontemporal_load(W4 + t);
    v4f p = P4[t];
    v4f ca = CA4[t];
    v4f cb = CB4[t];
    v4f r;
#pragma unroll
    for (int cidx = 0; cidx < 4; ++cidx) {
      // delta_t = f - p:  >0 -> potentiation, <0 -> depression, ==0 -> 0
      float dw = (f > p[cidx]) ? (rowP * ca[cidx])
                               : ((f < p[cidx]) ? (-rowD * cb[cidx]) : 0.f);
      r[cidx] = w[cidx] + dw;
    }
    __builtin_nontemporal_store(r, O4 + t);
  }
}

extern "C" void kernel_launch(void* const* d_in, const int* in_sizes, int n_in,
                              void* d_out, int out_size, void* d_ws, size_t ws_size,
                              hipStream_t stream) {
  (void)in_sizes; (void)n_in; (void)out_size; (void)ws_size;

  const float* in_spikes  = (const float*)d_in[0];
  const float* W          = (const float*)d_in[1];
  const float* membrane   = (const float*)d_in[2];
  const float* delta_pre  = (const float*)d_in[3];
  const float* delta_fire = (const float*)d_in[4];

  float* out = (float*)d_out;
  float* ws  = (float*)d_ws;
  float* weighted = ws;                 // [0,     8192)
  float* colA     = ws + OUT_F;         // [8192, 16384)
  float* colB     = ws + 2 * OUT_F;     // [16384,24576)

  // 1) GEVM: 512 workgroups x 16 rows, 8 waves each over K-slices.
  gevm_wmma_kernel<<<dim3(OUT_F / 16), dim3(256), 0, stream>>>(W, in_spikes, weighted);

  // 2) LIF + deltas + exp tables (tiny).
  lif_kernel<<<dim3(OUT_F / 256), dim3(256), 0, stream>>>(
      weighted, in_spikes, membrane, delta_pre, delta_fire, out, colA, colB);

  // 3) STDP weight update: one row per block, streaming read/modify/write.
  stdp_update_kernel<<<dim3(OUT_F), dim3(256), 0, stream>>>(
      W, out, colA, colB, out + 4 * OUT_F);
}